// NodeModel_43980465111676
// MI455X (gfx1250) — compile-verified
//
#include <hip/hip_runtime.h>

#define NN 40000
#define NE 640000
#define D  128

typedef __attribute__((ext_vector_type(2))) float v2f;
typedef __attribute__((ext_vector_type(8))) float v8f;

// ---------------------------------------------------------------- zero init
__global__ __launch_bounds__(256) void zero_kernel(float* __restrict__ agg,
                                                   float* __restrict__ cnt) {
    int i = blockIdx.x * 256 + threadIdx.x;
    if (i < NN * D) agg[i] = 0.0f;
    if (i < NN)     cnt[i] = 0.0f;
}

// ------------------------------------------------- fused 2-layer MLP (WMMA)
// out[n,128] = relu(in[n,K1] @ Wa[K1,128]) @ Wb[128,128]
// one block = 8 waves = 16 rows; wave w owns output cols [16w,16w+16)
template <int K1>
__global__ __launch_bounds__(256) void mlp2_kernel(const float* __restrict__ in,
                                                   const float* __restrict__ Wa,
                                                   const float* __restrict__ Wb,
                                                   float* __restrict__ out) {
    constexpr int IP = K1 + 4;   // padded LDS strides (avoid 16-way bank conflicts)
    constexpr int HP = D + 4;
    __shared__ float in_tile[16 * IP];
    __shared__ float h_tile[16 * HP];

    const int row0 = blockIdx.x * 16;
    const int tid  = threadIdx.x;

    // cooperative stage of the 16 x K1 input tile
    for (int idx = tid; idx < 16 * K1; idx += 256) {
        int r = idx / K1, c = idx - r * K1;
        in_tile[r * IP + c] = in[(row0 + r) * K1 + c];
    }
    __syncthreads();

    const int lane  = tid & 31;
    const int col0  = (tid >> 5) * 16;
    const int n     = lane & 15;        // B/D column, also A row (M)
    const int khalf = (lane >> 4) * 2;  // K sub-offset: 0 or 2
    const int mbase = (lane >> 4) * 8;  // D-matrix M base for this half-wave

    // ---- phase 1: h = relu(in @ Wa), K = K1
    v8f c1 = {};
    for (int k0 = 0; k0 < K1; k0 += 4) {
        const int kb = k0 + khalf;
        v2f a, b;
        a.x = in_tile[n * IP + kb];
        a.y = in_tile[n * IP + kb + 1];
        b.x = Wa[kb * D + col0 + n];
        b.y = Wa[(kb + 1) * D + col0 + n];
        c1 = __builtin_amdgcn_wmma_f32_16x16x4_f32(false, a, false, b,
                                                   (short)0, c1, false, false);
    }
#pragma unroll
    for (int r = 0; r < 8; ++r) {
        float v = c1[r];
        h_tile[(mbase + r) * HP + col0 + n] = v > 0.0f ? v : 0.0f;
    }
    __syncthreads();

    // ---- phase 2: out = h @ Wb, K = 128
    v8f c2 = {};
    for (int k0 = 0; k0 < D; k0 += 4) {
        const int kb = k0 + khalf;
        v2f a, b;
        a.x = h_tile[n * HP + kb];
        a.y = h_tile[n * HP + kb + 1];
        b.x = Wb[kb * D + col0 + n];
        b.y = Wb[(kb + 1) * D + col0 + n];
        c2 = __builtin_amdgcn_wmma_f32_16x16x4_f32(false, a, false, b,
                                                   (short)0, c2, false, false);
    }
#pragma unroll
    for (int r = 0; r < 8; ++r)
        out[(row0 + mbase + r) * D + col0 + n] = c2[r];
}

// ------------------------------------------------------ edge scatter (mean)
// one wave per edge: gather m[row] (L2-resident), atomic-add into agg[col]
__global__ __launch_bounds__(256) void scatter_kernel(const float4* __restrict__ m,
                                                      const int* __restrict__ ei,
                                                      float* __restrict__ agg,
                                                      float* __restrict__ cnt) {
    const int e    = (blockIdx.x * 256 + threadIdx.x) >> 5;
    const int lane = threadIdx.x & 31;
    if (e >= NE) return;
    const int row = ei[e];
    const int col = ei[NE + e];
    float4 v = m[row * (D / 4) + lane];
    float* dst = agg + col * D + lane * 4;
    atomicAdd(dst + 0, v.x);
    atomicAdd(dst + 1, v.y);
    atomicAdd(dst + 2, v.z);
    atomicAdd(dst + 3, v.w);
    if (lane == 0) atomicAdd(cnt + col, 1.0f);
}

// ------------------------------------------------ per-node layernorm fusion
__device__ __forceinline__ float wsum(float v) {
#pragma unroll
    for (int off = 16; off; off >>= 1) v += __shfl_xor(v, off, 32);
    return v;
}

__global__ __launch_bounds__(256) void fuse_kernel(const float* __restrict__ x,
                                                   const float* __restrict__ agg,
                                                   const float* __restrict__ cnt,
                                                   const float* __restrict__ g1p,
                                                   const float* __restrict__ b1p,
                                                   const float* __restrict__ g2p,
                                                   const float* __restrict__ b2p,
                                                   const float* __restrict__ wp,
                                                   float* __restrict__ cat) {
    const int node = (blockIdx.x * 256 + threadIdx.x) >> 5;
    const int lane = threadIdx.x & 31;
    if (node >= NN) return;
    const float inv = 1.0f / 128.0f;

    float sc = 1.0f / fmaxf(cnt[node], 1.0f);
    float4 a = ((const float4*)(agg + node * D))[lane];
    a.x *= sc; a.y *= sc; a.z *= sc; a.w *= sc;

    // LN1 on agg
    float mu = wsum(a.x + a.y + a.z + a.w) * inv;
    float dx = a.x - mu, dy = a.y - mu, dz = a.z - mu, dw = a.w - mu;
    float var = wsum(dx * dx + dy * dy + dz * dz + dw * dw) * inv;
    float r = rsqrtf(var + 1e-5f);
    float4 g1 = ((const float4*)g1p)[lane];
    float4 b1 = ((const float4*)b1p)[lane];
    float4 an;
    an.x = dx * r * g1.x + b1.x;  an.y = dy * r * g1.y + b1.y;
    an.z = dz * r * g1.z + b1.z;  an.w = dw * r * g1.w + b1.w;

    // repulsion + residual
    float4 xv = ((const float4*)(x + node * D))[lane];
    float4 wv = ((const float4*)wp)[lane];
    float4 y;
    y.x = xv.x + (xv.x - an.x) * wv.x;  y.y = xv.y + (xv.y - an.y) * wv.y;
    y.z = xv.z + (xv.z - an.z) * wv.z;  y.w = xv.w + (xv.w - an.w) * wv.w;

    // LN2 on y
    float mu2 = wsum(y.x + y.y + y.z + y.w) * inv;
    float ex = y.x - mu2, ey = y.y - mu2, ez = y.z - mu2, ew = y.w - mu2;
    float var2 = wsum(ex * ex + ey * ey + ez * ez + ew * ew) * inv;
    float r2 = rsqrtf(var2 + 1e-5f);
    float4 g2 = ((const float4*)g2p)[lane];
    float4 b2 = ((const float4*)b2p)[lane];
    float4 fx;
    fx.x = ex * r2 * g2.x + b2.x;  fx.y = ey * r2 * g2.y + b2.y;
    fx.z = ez * r2 * g2.z + b2.z;  fx.w = ew * r2 * g2.w + b2.w;

    ((float4*)(cat + node * 2 * D))[lane]      = fx;  // cols 0..127
    ((float4*)(cat + node * 2 * D + D))[lane]  = an;  // cols 128..255
}

// --------------------------------------------------------------- launcher
extern "C" void kernel_launch(void* const* d_in, const int* in_sizes, int n_in,
                              void* d_out, int out_size, void* d_ws, size_t ws_size,
                              hipStream_t stream) {
    const float* x   = (const float*)d_in[0];
    const int*   ei  = (const int*)d_in[1];
    const float* l1w = (const float*)d_in[2];
    const float* l1b = (const float*)d_in[3];
    const float* l2w = (const float*)d_in[4];
    const float* l2b = (const float*)d_in[5];
    const float* w   = (const float*)d_in[6];
    const float* W1  = (const float*)d_in[7];
    const float* W2  = (const float*)d_in[8];
    const float* O1  = (const float*)d_in[9];
    const float* O2  = (const float*)d_in[10];
    float* out = (float*)d_out;

    float* m   = (float*)d_ws;        // [NN, 128]
    float* agg = m + NN * D;          // [NN, 128]
    float* cnt = agg + NN * D;        // [NN]
    float* cat = cnt + NN;            // [NN, 256]   total ~82 MB

    zero_kernel<<<(NN * D + 255) / 256, 256, 0, stream>>>(agg, cnt);
    // per-NODE message MLP (16x less GEMM work than per-edge formulation)
    mlp2_kernel<128><<<NN / 16, 256, 0, stream>>>(x, W1, W2, m);
    // edge gather + atomic scatter (L2-resident working set)
    scatter_kernel<<<(NE * 32 + 255) / 256, 256, 0, stream>>>((const float4*)m, ei, agg, cnt);
    // mean + LN1 + repulsion + LN2 + concat
    fuse_kernel<<<(NN * 32 + 255) / 256, 256, 0, stream>>>(x, agg, cnt, l1w, l1b,
                                                           l2w, l2b, w, cat);
    // output MLP on [fx, agg_n]
    mlp2_kernel<256><<<NN / 16, 256, 0, stream>>>(cat, O1, O2, out);
}